// ConvAttn_90005334655730
// MI455X (gfx1250) — compile-verified
//
#include <hip/hip_runtime.h>
#include <hip/hip_bf16.h>
#include <math.h>

typedef __attribute__((ext_vector_type(2))) float v2f;
typedef __attribute__((ext_vector_type(8))) float v8f;

#define B_SZ     64
#define L_IN     4001
#define L_CONV   998
#define D_DIM    1000
#define KPAD     1024          // padded K stride for activations (zeros beyond 997)
#define NPAD     1008          // 63*16, padded N for packed weights
#define KP_W     1000          // packed-weight K rows (covers 250*4 loop, zeros beyond 997)
#define NT       63            // ceil(1000/16) N tiles
#define MT       4             // 64/16 M tiles
#define PK_MAT   (KP_W * NPAD) // 1,008,000 floats per packed matrix

// ---------------- workspace layout (floats) ----------------
#define WS_SCALE 0                       // 4096
#define WS_SHIFT 4096                    // 4096
#define WS_CONV  8192                    // 3 * 64 * 1024
#define WS_QKV   204800                  // 3 * 64 * 1024
#define WS_CTX   401408                  // 64 * 1024
#define WS_WT    466944                  // 4 * PK_MAT (q,k,v weights + out_w), K-major zero-padded
#define BR_STRIDE 65536                  // 64*1024 per branch

// ---------------- kernel 1: fold batchnorm ----------------
__global__ void bn_fold_kernel(const float* __restrict__ gamma,
                               const float* __restrict__ beta,
                               const float* __restrict__ mean,
                               const float* __restrict__ var,
                               float* __restrict__ ws) {
    int i = blockIdx.x * blockDim.x + threadIdx.x;
    if (i >= L_IN) return;
    float s = gamma[i] * rsqrtf(var[i] + 1e-5f);
    ws[WS_SCALE + i] = s;
    ws[WS_SHIFT + i] = beta[i] - mean[i] * s;
}

// ---------------- kernel 2: pack weights K-major with zero padding ----------------
// Wt[mat][k][n] = (valid) ? W[n][k] : 0 ;  mat 0..2 = lin_w_{q,k,v} (1000x998), mat 3 = out_w (1000x1000)
__global__ void pack_w_kernel(const float* __restrict__ lwq,
                              const float* __restrict__ lwk,
                              const float* __restrict__ lwv,
                              const float* __restrict__ oww,
                              float* __restrict__ ws) {
    int idx = blockIdx.x * blockDim.x + threadIdx.x;   // 4 * PK_MAT threads, exact grid
    int mat = idx / PK_MAT;
    int rem = idx - mat * PK_MAT;
    int k   = rem / NPAD;
    int n   = rem - k * NPAD;
    float v = 0.0f;
    if (mat < 3) {
        const float* W = (mat == 0) ? lwq : (mat == 1) ? lwk : lwv;
        if (n < D_DIM && k < L_CONV) v = W[n * L_CONV + k];
    } else {
        if (n < D_DIM) v = oww[n * D_DIM + k];          // k < 1000 always
    }
    ws[WS_WT + idx] = v;
}

// ---------------- kernel 3: fused BN + strided conv (3 branches) ----------------
__global__ void bn_conv_kernel(const float* __restrict__ x,
                               const float* __restrict__ cwq, const float* __restrict__ cbq,
                               const float* __restrict__ cwk, const float* __restrict__ cbk,
                               const float* __restrict__ cwv, const float* __restrict__ cbv,
                               float* __restrict__ ws) {
    int idx = blockIdx.x * blockDim.x + threadIdx.x;   // 3*64*1024 threads
    int branch = idx / BR_STRIDE;
    int rem    = idx - branch * BR_STRIDE;
    int b      = rem >> 10;
    int o      = rem & (KPAD - 1);
    float* conv = ws + WS_CONV;
    if (o >= L_CONV) { conv[idx] = 0.0f; return; }     // zero K padding for WMMA A-matrix
    const float* cw = (branch == 0) ? cwq : (branch == 1) ? cwk : cwv;
    float acc = (branch == 0) ? cbq[0] : (branch == 1) ? cbk[0] : cbv[0];
    const float* scale = ws + WS_SCALE;
    const float* shift = ws + WS_SHIFT;
    int base = 4 * o;
    const float* xr = x + b * L_IN + base;
#pragma unroll
    for (int t = 0; t < 10; ++t) {
        float xn = fmaf(xr[t], scale[base + t], shift[base + t]);
        acc = fmaf(xn, cw[t], acc);
    }
    conv[idx] = acc;
}

// ---------------- guard-free 16x16 WMMA tile over K=1000 ----------------
// A: activation rows (stride KPAD, zero-padded), Bp: packed K-major weights + nCol offset
__device__ __forceinline__ v8f wmma_tile_k1000(const float* __restrict__ A,
                                               const float* __restrict__ Bp,
                                               int half) {
    v8f acc = {0.f, 0.f, 0.f, 0.f, 0.f, 0.f, 0.f, 0.f};
#pragma unroll 5
    for (int ks = 0; ks < 250; ++ks) {
        int k0 = ks * 4 + half * 2;
        v2f a, b;
        a.x = A[k0];
        a.y = A[k0 + 1];
        b.x = Bp[k0 * NPAD];
        b.y = Bp[k0 * NPAD + NPAD];
        acc = __builtin_amdgcn_wmma_f32_16x16x4_f32(
            false, a, false, b, (short)0, acc, false, false);
    }
    return acc;
}

// ---------------- kernel 4: Q/K/V GEMM + bias + ReLU + in-proj affine ----------------
__global__ __launch_bounds__(128)
void qkv_gemm_kernel(const float* __restrict__ lbq,
                     const float* __restrict__ lbk,
                     const float* __restrict__ lbv,
                     const float* __restrict__ ipw, const float* __restrict__ ipb,
                     float* __restrict__ ws) {
    int wave   = (blockIdx.x * blockDim.x + threadIdx.x) >> 5;  // 0..755, exact grid
    int branch = wave / (MT * NT);
    int t      = wave - branch * (MT * NT);
    int mtile  = t & (MT - 1);
    int ntile  = t >> 2;
    int lane   = threadIdx.x & 31;
    int half   = lane >> 4;
    int l      = lane & 15;

    const float* A  = ws + WS_CONV + branch * BR_STRIDE + (mtile * 16 + l) * KPAD;
    int nCol        = ntile * 16 + l;                    // < 1008, pad cols are zero
    const float* Bp = ws + WS_WT + branch * PK_MAT + nCol;

    v8f acc = wmma_tile_k1000(A, Bp, half);

    if (nCol < D_DIM) {
        const float* Lb = (branch == 0) ? lbq : (branch == 1) ? lbk : lbv;
        float bias = Lb[nCol];
        float w    = ipw[branch];
        float bi   = ipb[branch];
        float* out = ws + WS_QKV + branch * BR_STRIDE;
#pragma unroll
        for (int r = 0; r < 8; ++r) {
            int brow = mtile * 16 + r + half * 8;        // D layout: VGPR r -> M=r (+8 upper half)
            float v = acc[r] + bias;
            v = v > 0.0f ? v : 0.0f;                     // ReLU
            out[brow * KPAD + nCol] = fmaf(v, w, bi);
        }
    }
}

// ---------------- kernel 5: rank-1 attention (softmax over 1000 keys) ----------------
__global__ __launch_bounds__(256)
void attn_kernel(const float* __restrict__ opw, const float* __restrict__ opb,
                 float* __restrict__ ws) {
    __shared__ float kl[D_DIM];
    __shared__ float vl[D_DIM];
    __shared__ float red[256];

    int b   = blockIdx.x;
    int tid = threadIdx.x;
    const float* q = ws + WS_QKV + 0 * BR_STRIDE + b * KPAD;
    const float* k = ws + WS_QKV + 1 * BR_STRIDE + b * KPAD;
    const float* v = ws + WS_QKV + 2 * BR_STRIDE + b * KPAD;

    float lmax = -3.4e38f, lmin = 3.4e38f;
    for (int j = tid; j < D_DIM; j += 256) {
        float kv = k[j];
        kl[j] = kv;
        vl[j] = v[j];
        lmax = fmaxf(lmax, kv);
        lmin = fminf(lmin, kv);
    }
    red[tid] = lmax; __syncthreads();
    for (int s = 128; s > 0; s >>= 1) { if (tid < s) red[tid] = fmaxf(red[tid], red[tid + s]); __syncthreads(); }
    float kmax = red[0]; __syncthreads();
    red[tid] = lmin; __syncthreads();
    for (int s = 128; s > 0; s >>= 1) { if (tid < s) red[tid] = fminf(red[tid], red[tid + s]); __syncthreads(); }
    float kmin = red[0]; __syncthreads();

    float ow = opw[0], ob = opb[0];
    float* ctx = ws + WS_CTX + b * KPAD;

    for (int it = 0; it < 4; ++it) {
        int i = it * 256 + tid;
        if (i < D_DIM) {
            float qi = q[i];
            float m  = (qi >= 0.0f) ? qi * kmax : qi * kmin;   // exact row max of rank-1 scores
            float s1 = 0.0f, s2 = 0.0f;
#pragma unroll 4
            for (int j = 0; j < D_DIM; ++j) {
                float e = __expf(fmaf(qi, kl[j], -m));         // v_exp_f32
                s1 += e;
                s2 = fmaf(e, vl[j], s2);
            }
            ctx[i] = fmaf(s2 / s1, ow, ob);
        }
    }
}

// ---------------- kernel 6: final GEMM  out = ctx @ out_w^T + out_b ----------------
__global__ __launch_bounds__(128)
void out_gemm_kernel(const float* __restrict__ obias,
                     const float* __restrict__ ws, float* __restrict__ out) {
    int wave  = (blockIdx.x * blockDim.x + threadIdx.x) >> 5;  // 0..251, exact grid
    int mtile = wave & (MT - 1);
    int ntile = wave >> 2;
    int lane  = threadIdx.x & 31;
    int half  = lane >> 4;
    int l     = lane & 15;

    const float* A  = ws + WS_CTX + (mtile * 16 + l) * KPAD;
    int nCol        = ntile * 16 + l;
    const float* Bp = ws + WS_WT + 3 * PK_MAT + nCol;

    v8f acc = wmma_tile_k1000(A, Bp, half);

    if (nCol < D_DIM) {
        float bias = obias[nCol];
#pragma unroll
        for (int r = 0; r < 8; ++r) {
            int brow = mtile * 16 + r + half * 8;
            out[brow * D_DIM + nCol] = acc[r] + bias;
        }
    }
}

extern "C" void kernel_launch(void* const* d_in, const int* in_sizes, int n_in,
                              void* d_out, int out_size, void* d_ws, size_t ws_size,
                              hipStream_t stream) {
    const float* x        = (const float*)d_in[0];
    const float* bn_gamma = (const float*)d_in[1];
    const float* bn_beta  = (const float*)d_in[2];
    const float* bn_mean  = (const float*)d_in[3];
    const float* bn_var   = (const float*)d_in[4];
    const float* cwq = (const float*)d_in[5];
    const float* cbq = (const float*)d_in[6];
    const float* lwq = (const float*)d_in[7];
    const float* lbq = (const float*)d_in[8];
    const float* cwk = (const float*)d_in[9];
    const float* cbk = (const float*)d_in[10];
    const float* lwk = (const float*)d_in[11];
    const float* lbk = (const float*)d_in[12];
    const float* cwv = (const float*)d_in[13];
    const float* cbv = (const float*)d_in[14];
    const float* lwv = (const float*)d_in[15];
    const float* lbv = (const float*)d_in[16];
    const float* ipw = (const float*)d_in[17];  // (3,1)
    const float* ipb = (const float*)d_in[18];  // (3,)
    const float* opw = (const float*)d_in[19];  // (1,1)
    const float* opb = (const float*)d_in[20];  // (1,)
    const float* oww = (const float*)d_in[21];  // (1000,1000)
    const float* owb = (const float*)d_in[22];  // (1000,)
    float* out = (float*)d_out;                 // (64,1,1000)
    float* ws  = (float*)d_ws;

    // 1) fold BN
    bn_fold_kernel<<<(L_IN + 255) / 256, 256, 0, stream>>>(bn_gamma, bn_beta, bn_mean, bn_var, ws);
    // 2) pack weights K-major, zero-padded (guard-free WMMA loops)
    pack_w_kernel<<<(4 * PK_MAT) / 256, 256, 0, stream>>>(lwq, lwk, lwv, oww, ws);
    // 3) BN + conv, all 3 branches, zero-padded K
    bn_conv_kernel<<<(3 * B_SZ * KPAD) / 256, 256, 0, stream>>>(x, cwq, cbq, cwk, cbk, cwv, cbv, ws);
    // 4) Q/K/V GEMM via WMMA: 756 tiles, 4 waves/block
    qkv_gemm_kernel<<<756 / 4, 128, 0, stream>>>(lbq, lbk, lbv, ipw, ipb, ws);
    // 5) attention, one block per batch
    attn_kernel<<<B_SZ, 256, 0, stream>>>(opw, opb, ws);
    // 6) final GEMM via WMMA: 252 tiles, 4 waves/block
    out_gemm_kernel<<<252 / 4, 128, 0, stream>>>(owb, ws, out);
}